// KSOM_64080912056524
// MI455X (gfx1250) — compile-verified
//
#include <hip/hip_runtime.h>

// Kohonen SOM update for MI455X (gfx1250, wave32).
// O = 512 (rows == batch), D = 128. All fp32.
//
// Decomposition:
//   dist[i,j]   = sum_d (W[i,j,d]-x[i,d])^2          (pass over W, 128 MB)
//   winner[i]   = argmin_j dist[i,j]                  (packed u64 atomicMin)
//   nb          = exp(-dist / (2*sigma^2)),  sigma=O/2
//   colsum[j,d] = LR * ( (nb^T x)[j,d] - sum_i nb[i,j]*W[i,j,d] )
//   out[i,j,d]  = W[i,j,d] + colsum[j,d]
//
// Roofline: HBM-bound. W = 128 MB fits MI455X's 192 MB L2, so passes 2/3
// re-read it from L2; the 128 MB output stream is written NON-TEMPORAL so it
// does not evict W from L2 (CDNA5 store TH=NT). nb^T x is a real
// 512x128x512 GEMM -> V_WMMA_F32_16X16X4_F32 (fp32 matrix path; bandwidth-
// bound problem, so no reason to drop precision).

typedef float v2f __attribute__((ext_vector_type(2)));
typedef float v4f __attribute__((ext_vector_type(4)));
typedef float v8f __attribute__((ext_vector_type(8)));

#define SOM_O 512
#define SOM_D 128
#define SOM_LR 0.01f
// sigma = O/2 = 256 -> 2*sigma^2 = 131072
#define SOM_INV2SIG2 (1.0f / 131072.0f)

// ---------------------------------------------------------------- K0: init
__global__ void som_init(unsigned long long* __restrict__ wpack) {
    int i = threadIdx.x;
    if (i < SOM_O) wpack[i] = 0xFFFFFFFFFFFFFFFFULL;
}

// ------------------------------------------- K1: distances + nb + argmin key
// One wave32 per (i,j) pair. Lane l loads float4 covering d = 4l..4l+3.
__global__ __launch_bounds__(256) void som_dist(const float* __restrict__ x,
                                                const float* __restrict__ w,
                                                float* __restrict__ nb,
                                                unsigned long long* __restrict__ wpack) {
    const int lane = threadIdx.x & 31;
    const int gw   = blockIdx.x * 8 + (threadIdx.x >> 5);   // 0 .. O*O-1
    const int i    = gw >> 9;
    const int j    = gw & (SOM_O - 1);

    const v4f* w4 = (const v4f*)(w + (size_t)gw * SOM_D);
    const v4f* x4 = (const v4f*)(x + (size_t)i * SOM_D);
    v4f wv = w4[lane];
    v4f xv = x4[lane];
    v4f d  = wv - xv;
    float s = d.x * d.x + d.y * d.y + d.z * d.z + d.w * d.w;

#pragma unroll
    for (int off = 16; off > 0; off >>= 1)
        s += __shfl_xor(s, off, 32);                        // wave32 reduction

    if (lane == 0) {
        nb[gw] = __expf(-s * SOM_INV2SIG2);
        // dist >= 0 so float bit pattern preserves ordering; low 32 bits = j
        // so ties resolve to the smallest j (matches argmin-first semantics).
        unsigned long long key =
            ((unsigned long long)__float_as_uint(s) << 32) | (unsigned)j;
        atomicMin(&wpack[i], key);
    }
}

// ------------------------------------------------- K2: G = nb^T @ x (WMMA)
// One wave per 16x16 output tile of G (O x D). K runs over i in steps of 4.
// Fragment layouts per CDNA5 ISA 7.12.2 (fp32 16x16x4):
//   A 16x4:  lanes 0-15 hold M=lane, K={0,1}; lanes 16-31 hold M=lane-16, K={2,3}
//   B 4x16:  mirrored (VGPR = K within half, lane-half selects K offset 2)
//   C/D:     VGPR r -> M = r + 8*(lane>=16), N = lane&15
__global__ __launch_bounds__(32) void som_gemm(const float* __restrict__ nb,
                                               const float* __restrict__ x,
                                               float* __restrict__ G) {
    const int lane = threadIdx.x;        // 0..31
    const int m    = lane & 15;
    const int half = lane >> 4;          // 0 or 1
    const int jt   = blockIdx.x >> 3;    // 0..31  (O/16 tiles)
    const int dt   = blockIdx.x & 7;     // 0..7   (D/16 tiles)
    const int j0   = jt * 16;
    const int d0   = dt * 16;

    v8f c = {};
    for (int i0 = 0; i0 < SOM_O; i0 += 4) {
        const int k0 = i0 + 2 * half;
        v2f a, b;
        // A[m][k] = nb^T[j0+m][k] = nb[k*O + j0+m]
        a.x = nb[(size_t)(k0 + 0) * SOM_O + (j0 + m)];
        a.y = nb[(size_t)(k0 + 1) * SOM_O + (j0 + m)];
        // B[k][n] = x[k*D + d0+n], n = lane&15
        b.x = x[(size_t)(k0 + 0) * SOM_D + (d0 + m)];
        b.y = x[(size_t)(k0 + 1) * SOM_D + (d0 + m)];
        c = __builtin_amdgcn_wmma_f32_16x16x4_f32(
                /*neg_a=*/false, a, /*neg_b=*/false, b,
                /*c_mod=*/(short)0, c, /*reuse_a=*/false, /*reuse_b=*/false);
    }

#pragma unroll
    for (int r = 0; r < 8; ++r) {
        const int mo = r + 8 * half;
        G[(size_t)(j0 + mo) * SOM_D + (d0 + m)] = c[r];
    }
}

// ------------------------- K3: colsum[j,d] = LR*(G[j,d] - sum_i nb[i,j]*W[i,j,d])
// One block per j, thread d. Sequential i-loop -> bit-deterministic.
// W re-read should hit L2 (kept warm by K1, not evicted thanks to NT output).
__global__ __launch_bounds__(128) void som_colsum(const float* __restrict__ nb,
                                                  const float* __restrict__ w,
                                                  const float* __restrict__ G,
                                                  float* __restrict__ colsum) {
    const int j = blockIdx.x;
    const int d = threadIdx.x;
    float acc = 0.0f;
    for (int i = 0; i < SOM_O; ++i) {
        const size_t ij = (size_t)i * SOM_O + j;
        acc = fmaf(nb[ij], w[ij * SOM_D + d], acc);
    }
    const int jd = j * SOM_D + d;
    colsum[jd] = SOM_LR * (G[jd] - acc);
}

// ---------------------------------------------------------- K4: winners out
__global__ void som_winner(const unsigned long long* __restrict__ wpack,
                           int* __restrict__ winner) {
    int i = blockIdx.x * blockDim.x + threadIdx.x;
    if (i < SOM_O) winner[i] = (int)(unsigned)(wpack[i] & 0xFFFFFFFFULL);
}

// --------------------------------- K5: out = W + broadcast(colsum), float4
// Output is write-once / never re-read: non-temporal stores keep the 128 MB
// result stream from evicting W out of the 192 MB L2.
__global__ __launch_bounds__(256) void som_add(const float* __restrict__ w,
                                               const float* __restrict__ colsum,
                                               float* __restrict__ out) {
    const size_t total4 = (size_t)SOM_O * SOM_O * SOM_D / 4;  // 8,388,608
    const v4f* w4 = (const v4f*)w;
    const v4f* c4 = (const v4f*)colsum;
    v4f*       o4 = (v4f*)out;
    for (size_t f = (size_t)blockIdx.x * blockDim.x + threadIdx.x; f < total4;
         f += (size_t)gridDim.x * blockDim.x) {
        const size_t e   = f * 4;
        const size_t jd4 = (e & (size_t)(SOM_O * SOM_D - 1)) >> 2;
        v4f r = w4[f] + c4[jd4];
        __builtin_nontemporal_store(r, &o4[f]);   // global_store_b128, TH=NT
    }
}

extern "C" void kernel_launch(void* const* d_in, const int* in_sizes, int n_in,
                              void* d_out, int out_size, void* d_ws, size_t ws_size,
                              hipStream_t stream) {
    (void)in_sizes; (void)n_in; (void)out_size; (void)ws_size;
    const float* x = (const float*)d_in[0];            // (O, D)
    const float* w = (const float*)d_in[1];            // (O, O, D)

    // Workspace layout (fp32 unless noted); total ~1.6 MB.
    float* nb     = (float*)d_ws;                      // O*O
    float* G      = nb + (size_t)SOM_O * SOM_O;        // O*D
    float* colsum = G + (size_t)SOM_O * SOM_D;         // O*D
    unsigned long long* wpack =
        (unsigned long long*)(colsum + (size_t)SOM_O * SOM_D);  // O (8B aligned)

    int*   outWinner = (int*)d_out;                    // first O slots (int32)
    float* outW      = (float*)d_out + SOM_O;          // O*O*D floats

    som_init  <<<1, SOM_O, 0, stream>>>(wpack);
    som_dist  <<<(SOM_O * SOM_O) / 8, 256, 0, stream>>>(x, w, nb, wpack);
    som_gemm  <<<(SOM_O / 16) * (SOM_D / 16), 32, 0, stream>>>(nb, x, G);
    som_colsum<<<SOM_O, SOM_D, 0, stream>>>(nb, w, G, colsum);
    som_winner<<<2, 256, 0, stream>>>(wpack, outWinner);
    som_add   <<<4096, 256, 0, stream>>>(w, colsum, outW);
}